// BoundConvexViolationProjection_35648228557557
// MI455X (gfx1250) — compile-verified
//
#include <hip/hip_runtime.h>
#include <math.h>

// BoundConvexViolationProjection on gfx1250.
// One workgroup per (b,s) pair; full iterative loop fused in-kernel.
// residual = A@x via v_wmma_f32_16x16x4_f32 (16-row tiles per wave, x
// replicated across B columns so any D column is the dot product);
// penalty_grad = A^T@viol via coalesced float2 column sweep + prefetch.

typedef __attribute__((ext_vector_type(2))) float v2f;
typedef __attribute__((ext_vector_type(8))) float v8f;

#define KLR       0.005f
#define KDELTA    0.1f
#define KMAX_ITER 100
#define KEPS      1e-6f
#define DN        512
#define DM        512

__launch_bounds__(256, 2)
__global__ void bcvp_kernel(const float* __restrict__ X,
                            const float* __restrict__ A,
                            const float* __restrict__ Bv,
                            const float* __restrict__ VM,
                            float* __restrict__ Out,
                            int S)
{
    __shared__ __align__(16) float s_x[DN];     // raw x
    __shared__ __align__(16) float s_xm[DN];    // x * var_mask
    __shared__ __align__(16) float s_vm[DN];    // var_mask
    __shared__ __align__(16) float s_b[DM];     // b vector
    __shared__ __align__(16) float s_cm[DM];    // constraint mask (0/1)
    __shared__ __align__(16) float s_viol[DM];  // residual -> violation
    __shared__ float s_wred[8];
    __shared__ float s_scalar;

    const int tid  = threadIdx.x;
    const int lane = tid & 31;
    const int wave = tid >> 5;
    const int blk  = blockIdx.x;      // b*S + s
    const int bidx = blk / S;

    const float* Ab = A  + (size_t)blk  * DM * DN;
    const float* bp = Bv + (size_t)blk  * DM;
    const float* xp = X  + (size_t)blk  * DN;
    const float* vp = VM + (size_t)bidx * DN;
    float*       op = Out + (size_t)blk * DN;

    // ---- load vectors into LDS ----
    for (int n = tid; n < DN; n += 256) {
        float v = vp[n], x = xp[n];
        s_vm[n] = v; s_x[n] = x; s_xm[n] = x * v;
    }
    for (int m = tid; m < DM; m += 256) s_b[m] = bp[m];
    __syncthreads();

    // ---- constraint mask: cm[m] = (sum_n vm[n]*|A[m,n]| > 0) ----
    for (int m = wave; m < DM; m += 8) {
        const float* row = Ab + (size_t)m * DN;
        float acc = 0.f;
        for (int k = lane; k < DN; k += 32)
            acc += fabsf(row[k]) * s_vm[k];
        for (int off = 16; off > 0; off >>= 1)
            acc += __shfl_xor(acc, off, 32);
        if (lane == 0) s_cm[m] = (acc > 0.f) ? 1.f : 0.f;
    }
    __syncthreads();

    // ISA 16x4 f32 A-frag layout: lanes0-15 hold K0(v0),K1(v1); lanes16-31 K2,K3.
    const int rowOfLane = lane & 15;
    const int kofs      = (lane >> 4) << 1;

    for (int it = 0; it < KMAX_ITER; ++it) {
        // ---- pass 1: residual_raw = A @ xm via WMMA (16 rows per wave/tile) ----
        for (int t = 0; t < 4; ++t) {
            const int m0 = t * 128 + wave * 16;
            const float* rp = Ab + (size_t)(m0 + rowOfLane) * DN + kofs;
            v8f acc = {0.f,0.f,0.f,0.f,0.f,0.f,0.f,0.f};
            #pragma unroll 4
            for (int k = 0; k < DN; k += 4) {
                v2f av = *(const v2f*)(rp + k);           // A-matrix fragment
                v2f bv = *(const v2f*)(&s_xm[k + kofs]);  // B: x chunk, same in all 16 cols
                acc = __builtin_amdgcn_wmma_f32_16x16x4_f32(
                        false, av, false, bv, (short)0, acc, false, false);
            }
            // C/D layout: vgpr r: lanes0-15 -> M=r, lanes16-31 -> M=r+8 (all cols equal)
            if (lane == 0) {
                #pragma unroll
                for (int r = 0; r < 8; ++r) s_viol[m0 + r] = acc[r];
            } else if (lane == 16) {
                #pragma unroll
                for (int r = 0; r < 8; ++r) s_viol[m0 + 8 + r] = acc[r];
            }
        }
        __syncthreads();

        // ---- violation + total violation ----
        float tvp = 0.f;
        for (int m = tid; m < DM; m += 256) {
            float r   = (s_viol[m] - s_b[m]) * s_cm[m];
            float pos = fmaxf(r, 0.f);
            float neg = fmaxf(-r - KDELTA, 0.f);
            tvp += pos;
            s_viol[m] = pos - neg;
        }
        for (int off = 16; off > 0; off >>= 1)
            tvp += __shfl_xor(tvp, off, 32);
        if (lane == 0) s_wred[wave] = tvp;
        __syncthreads();
        if (tid == 0) {
            float tv = 0.f;
            #pragma unroll
            for (int w = 0; w < 8; ++w) tv += s_wred[w];
            s_scalar = tv;
        }
        __syncthreads();
        const bool active = (s_scalar >= KDELTA);
        __syncthreads();              // protect s_scalar/s_wred reuse
        if (!active) break;           // uniform: pair frozen, matches reference

        // ---- pass 2: penalty_grad = A^T @ viol (2 cols/thread, coalesced) ----
        const int n0 = tid << 1;
        const float* cp = Ab + n0;
        float pg0 = 0.f, pg1 = 0.f;
        for (int m = 0; m < DM; m += 8) {
            int pm = m + 16; if (pm >= DM) pm = DM - 1;
            __builtin_prefetch(cp + (size_t)pm * DN, 0, 0);
            #pragma unroll
            for (int mm = 0; mm < 8; ++mm) {
                float v = s_viol[m + mm];
                v2f a2 = *(const v2f*)(cp + (size_t)(m + mm) * DN);
                pg0 = fmaf(v, a2.x, pg0);
                pg1 = fmaf(v, a2.y, pg1);
            }
        }
        float np = pg0 * pg0 + pg1 * pg1;
        for (int off = 16; off > 0; off >>= 1)
            np += __shfl_xor(np, off, 32);
        if (lane == 0) s_wred[wave] = np;
        __syncthreads();
        if (tid == 0) {
            float nrm = 0.f;
            #pragma unroll
            for (int w = 0; w < 8; ++w) nrm += s_wred[w];
            s_scalar = sqrtf(nrm) + KEPS;
        }
        __syncthreads();
        const float step = KLR / s_scalar;
        float x0 = fmaxf(s_x[n0]     - step * pg0, 0.f);
        float x1 = fmaxf(s_x[n0 + 1] - step * pg1, 0.f);
        s_x[n0]      = x0;  s_x[n0 + 1]  = x1;
        s_xm[n0]     = x0 * s_vm[n0];
        s_xm[n0 + 1] = x1 * s_vm[n0 + 1];
        __syncthreads();
    }

    for (int n = tid; n < DN; n += 256)
        op[n] = s_x[n] * s_vm[n];
}

extern "C" void kernel_launch(void* const* d_in, const int* in_sizes, int n_in,
                              void* d_out, int out_size, void* d_ws, size_t ws_size,
                              hipStream_t stream) {
    (void)n_in; (void)out_size; (void)d_ws; (void)ws_size;
    const float* x  = (const float*)d_in[0];   // (B,S,N)
    const float* A  = (const float*)d_in[1];   // (B,S,M,N)
    const float* b  = (const float*)d_in[2];   // (B,S,M)
    const float* vm = (const float*)d_in[3];   // (B,N)
    float* out = (float*)d_out;                // (B,S,N)

    const int BS = in_sizes[2] / DM;   // B*S
    const int Bb = in_sizes[3] / DN;   // B
    const int S  = BS / Bb;

    bcvp_kernel<<<BS, 256, 0, stream>>>(x, A, b, vm, out, S);
}